// GCNEdge_77335181132320
// MI455X (gfx1250) — compile-verified
//
#include <hip/hip_runtime.h>
#include <hip/hip_bf16.h>

#define N_NODES 50000
#define N_EDGES 800000
#define IN_FEATS 128
#define HIDDEN 64
#define M_TILES (N_NODES / 16)   // 3125, exact

typedef float v2f __attribute__((ext_vector_type(2)));
typedef float v8f __attribute__((ext_vector_type(8)));

// ---------------- degree counting ----------------
__global__ void count_kernel(const int* __restrict__ src, const int* __restrict__ dst,
                             int* __restrict__ cnt_src, int* __restrict__ cnt_dst, int E) {
    int e = blockIdx.x * blockDim.x + threadIdx.x;
    if (e >= E) return;
    atomicAdd(&cnt_src[src[e]], 1);
    atomicAdd(&cnt_dst[dst[e]], 1);
}

// counts (int) -> c = rsqrt(max(deg,1)) (float), in place
__global__ void coeff_kernel(int* __restrict__ cs, int* __restrict__ cd, int n) {
    int i = blockIdx.x * blockDim.x + threadIdx.x;
    if (i >= n) return;
    int a = cs[i]; int b = cd[i];
    ((float*)cs)[i] = rsqrtf((float)(a < 1 ? 1 : a));
    ((float*)cd)[i] = rsqrtf((float)(b < 1 ? 1 : b));
}

// ---------------- WMMA GEMM: out = A @ W ----------------
// A: N_NODES x K (row major), W: K x 64 (row major), out: N_NODES x 64.
// W staged in LDS pre-paired along K: sB[j*64+n] = {W[2j][n], W[2j+1][n]} so every
// WMMA B operand is one aligned ds_load_b64 into an even VGPR pair (no shuffles).
// One wave computes a 16x64 strip with 4 f32 accumulators (V_WMMA_F32_16X16X4_F32).
template<int K>
__global__ __launch_bounds__(128) void gemm_wmma_kernel(
        const float* __restrict__ A, const float* __restrict__ W,
        float* __restrict__ out, int mtiles) {
    __shared__ v2f sB[(K / 2) * 64];
    for (int i = (int)threadIdx.x; i < (K / 2) * 64; i += 128) {
        const int j = i >> 6, n = i & 63;
        v2f p;
        p.x = W[(2 * j)     * 64 + n];
        p.y = W[(2 * j + 1) * 64 + n];
        sB[i] = p;
    }
    __syncthreads();

    const int lane  = (int)threadIdx.x & 31;
    const int mtile = (int)blockIdx.x * 4 + ((int)threadIdx.x >> 5);
    if (mtile >= mtiles) return;                 // wave-uniform exit: EXEC all-1 for WMMA
    const int m_base = mtile * 16;
    const int col    = lane & 15;
    const int half   = lane >> 4;                // 0: K=k0,k0+1   1: K=k0+2,k0+3
    const int row    = m_base + col;             // A-layout: both halves hold M=0..15
    const float* __restrict__ arow = A + (size_t)row * K;

    v8f acc0 = {}, acc1 = {}, acc2 = {}, acc3 = {};
    for (int k0 = 0; k0 < K; k0 += 4) {
        const int kb = k0 + (half << 1);                       // even
        const v2f a = *(const v2f*)(arow + kb);                // global_load_b64
        const v2f* __restrict__ brow = sB + ((k0 >> 1) + half) * 64;
        const v2f b0 = brow[     col];                         // ds_load_b64 each
        const v2f b1 = brow[16 + col];
        const v2f b2 = brow[32 + col];
        const v2f b3 = brow[48 + col];
        acc0 = __builtin_amdgcn_wmma_f32_16x16x4_f32(false, a, false, b0, (short)0, acc0, false, false);
        acc1 = __builtin_amdgcn_wmma_f32_16x16x4_f32(false, a, false, b1, (short)0, acc1, false, false);
        acc2 = __builtin_amdgcn_wmma_f32_16x16x4_f32(false, a, false, b2, (short)0, acc2, false, false);
        acc3 = __builtin_amdgcn_wmma_f32_16x16x4_f32(false, a, false, b3, (short)0, acc3, false, false);
    }

    // C/D layout: VGPR v -> M = v (lanes 0-15) or v+8 (lanes 16-31); N = lane&15
    const int rtop = half << 3;
    #pragma unroll
    for (int v = 0; v < 8; ++v) {
        const size_t r = (size_t)(m_base + rtop + v) * 64;
        out[r +      col] = acc0[v];
        out[r + 16 + col] = acc1[v];
        out[r + 32 + col] = acc2[v];
        out[r + 48 + col] = acc3[v];
    }
}

// ---------------- edge scatter: agg[dst] += y[src] * c_src[src] (64 feats) ----------------
__global__ void scatter64_kernel(const float* __restrict__ y, const float* __restrict__ csrc,
                                 const int* __restrict__ src, const int* __restrict__ dst,
                                 float* __restrict__ agg, int E) {
    int e = blockIdx.x * 4 + ((int)threadIdx.x >> 6);
    int f = (int)threadIdx.x & 63;
    if (e >= E) return;
    int s = src[e], d = dst[e];
    float v = y[(size_t)s * 64 + f] * csrc[s];
    atomicAdd(&agg[(size_t)d * 64 + f], v);
}

// ---------------- epilogue: h = act(agg * c_dst[row] + bias[col]), in place ----------------
template<bool RELU>
__global__ void epilogue_kernel(float* __restrict__ agg, const float* __restrict__ cdst,
                                const float* __restrict__ bias, int n_elems) {
    int i = blockIdx.x * blockDim.x + threadIdx.x;
    if (i >= n_elems) return;
    int row = i >> 6, f = i & 63;
    float v = agg[i] * cdst[row] + bias[f];
    if (RELU) v = fmaxf(v, 0.0f);
    agg[i] = v;
}

// ---------------- edge scoring: sigmoid(h[s]@Ws + h[d]@Wd + ef@Wf + be) ----------------
__global__ void edge_kernel(const float* __restrict__ h, const int* __restrict__ src,
                            const int* __restrict__ dst, const float* __restrict__ efeat,
                            const float* __restrict__ We, const float* __restrict__ be,
                            float* __restrict__ out, int E) {
    const int lane = (int)threadIdx.x & 31;
    const int e = (int)blockIdx.x * ((int)blockDim.x >> 5) + ((int)threadIdx.x >> 5);
    if (e >= E) return;
    const int s = src[e], d = dst[e];
    const float* hs = h + (size_t)s * 64;
    const float* hd = h + (size_t)d * 64;
    float sum = hs[lane] * We[lane] + hs[32 + lane] * We[32 + lane]
              + hd[lane] * We[64 + lane] + hd[32 + lane] * We[96 + lane];
    #pragma unroll
    for (int off = 16; off > 0; off >>= 1) sum += __shfl_xor(sum, off, 32);
    if (lane == 0) {
        sum += efeat[(size_t)e * 3 + 0] * We[128]
             + efeat[(size_t)e * 3 + 1] * We[129]
             + efeat[(size_t)e * 3 + 2] * We[130] + be[0];
        out[e] = 1.0f / (1.0f + expf(-sum));
    }
}

extern "C" void kernel_launch(void* const* d_in, const int* in_sizes, int n_in,
                              void* d_out, int out_size, void* d_ws, size_t ws_size,
                              hipStream_t stream) {
    const float* x     = (const float*)d_in[0];
    const float* efeat = (const float*)d_in[1];
    const int*   src   = (const int*)  d_in[2];
    const int*   dst   = (const int*)  d_in[3];
    const float* W1    = (const float*)d_in[4];
    const float* b1    = (const float*)d_in[5];
    const float* W2    = (const float*)d_in[6];
    const float* b2    = (const float*)d_in[7];
    const float* We    = (const float*)d_in[8];
    const float* be    = (const float*)d_in[9];
    float* out = (float*)d_out;

    const size_t NB = (size_t)N_NODES * 64;      // elements per node-feature buffer
    char* ws = (char*)d_ws;
    float* bufA = (float*)ws;                    // y1, then (zeroed) agg2 -> h2
    float* bufB = bufA + NB;                     // agg1 -> h1 (in place)
    float* bufC = bufB + NB;                     // y2
    int*   cntS = (int*)(bufC + NB);             // N ints -> c_src (in place)
    int*   cntD = cntS + N_NODES;                // N ints -> c_dst (in place)
    float* cS   = (float*)cntS;
    float* cD   = (float*)cntD;

    hipMemsetAsync(cntS, 0, (size_t)2 * N_NODES * sizeof(int), stream);
    hipMemsetAsync(bufB, 0, NB * sizeof(float), stream);

    count_kernel<<<N_EDGES / 256, 256, 0, stream>>>(src, dst, cntS, cntD, N_EDGES);
    coeff_kernel<<<(N_NODES + 255) / 256, 256, 0, stream>>>(cntS, cntD, N_NODES);

    // layer 1: y1 = x @ W1 ; agg1[dst] += c_src[s]*y1[s] ; h1 = relu(c_dst*agg1 + b1)
    gemm_wmma_kernel<128><<<(M_TILES + 3) / 4, 128, 0, stream>>>(x, W1, bufA, M_TILES);
    scatter64_kernel<<<N_EDGES / 4, 256, 0, stream>>>(bufA, cS, src, dst, bufB, N_EDGES);
    epilogue_kernel<true><<<(int)(NB / 256), 256, 0, stream>>>(bufB, cD, b1, (int)NB);

    // layer 2: y2 = h1 @ W2 ; agg2[dst] += c_src[s]*y2[s] ; h2 = c_dst*agg2 + b2
    gemm_wmma_kernel<64><<<(M_TILES + 3) / 4, 128, 0, stream>>>(bufB, W2, bufC, M_TILES);
    hipMemsetAsync(bufA, 0, NB * sizeof(float), stream);
    scatter64_kernel<<<N_EDGES / 4, 256, 0, stream>>>(bufC, cS, src, dst, bufA, N_EDGES);
    epilogue_kernel<false><<<(int)(NB / 256), 256, 0, stream>>>(bufA, cD, b2, (int)NB);

    // edge scores
    edge_kernel<<<N_EDGES / 8, 256, 0, stream>>>(bufA, src, dst, efeat, We, be, out, N_EDGES);
}